// PointGeneratorGCN_47545287967252
// MI455X (gfx1250) — compile-verified
//
#include <hip/hip_runtime.h>

typedef float v2f __attribute__((ext_vector_type(2)));
typedef float v8f __attribute__((ext_vector_type(8)));

static constexpr int N_NODES  = 100000;
static constexpr int N_EDGES  = 3200000;
static constexpr int IN_DIM   = 64;
static constexpr int HID      = 128;
static constexpr int OUT_DIM  = 32;
static constexpr int N_GRAPHS = 64;

__device__ __forceinline__ void atomic_add_f32(float* p, float v) {
    __hip_atomic_fetch_add(p, v, __ATOMIC_RELAXED, __HIP_MEMORY_SCOPE_AGENT);
}

// ---------------------------------------------------------------------------
// Degree / normalization
// ---------------------------------------------------------------------------
__global__ void k_deg_init(float* __restrict__ deg, int n) {
    int i = blockIdx.x * blockDim.x + threadIdx.x;
    if (i < n) deg[i] = 1.0f;               // self-loop contributes 1
}

__global__ void k_deg_count(const int* __restrict__ col, float* __restrict__ deg, int nE) {
    int i = blockIdx.x * blockDim.x + threadIdx.x;
    if (i < nE) atomic_add_f32(&deg[col[i]], 1.0f);
}

__global__ void k_dinv(const float* __restrict__ deg, float* __restrict__ dinv, int n) {
    int i = blockIdx.x * blockDim.x + threadIdx.x;
    if (i < n) dinv[i] = rsqrtf(deg[i]);    // deg >= 1 always (self loops)
}

__global__ void k_zero_pool(float* __restrict__ sums, float* __restrict__ counts) {
    int i = blockIdx.x * blockDim.x + threadIdx.x;
    if (i < N_GRAPHS * HID) sums[i] = 0.0f;
    if (i < N_GRAPHS)       counts[i] = 0.0f;
}

// ---------------------------------------------------------------------------
// WMMA GEMM: C[M x (NTILES*16)] = op(A)[M x K] * B[K x (NTILES*16)]  (+ bias)
// One wave32 computes a 16-row strip across all N columns using
// V_WMMA_F32_16X16X4_F32. A-frag: lanes 0-15 hold K=0,1; lanes 16-31 K=2,3.
// ---------------------------------------------------------------------------
template <int NTILES, bool RELU_A, bool BIAS>
__global__ __launch_bounds__(256)
void k_wmma_gemm(const float* __restrict__ A, const float* __restrict__ B,
                 const float* __restrict__ bias, float* __restrict__ C,
                 int M, int K) {
    const int N    = NTILES * 16;
    int wave = (blockIdx.x * blockDim.x + threadIdx.x) >> 5;
    int lane = threadIdx.x & 31;
    int row0 = wave << 4;
    if (row0 >= M) return;                  // wave-uniform: EXEC stays all-1s
    int half = lane >> 4;                   // 0: K=0,1   1: K=2,3
    int l16  = lane & 15;

    v8f acc[NTILES] = {};

    const float* __restrict__ arow = A + (size_t)(row0 + l16) * K + 2 * half;
    for (int k = 0; k < K; k += 4) {
        float2 av = *(const float2*)(arow + k);
        if (RELU_A) { av.x = fmaxf(av.x, 0.0f); av.y = fmaxf(av.y, 0.0f); }
        v2f a; a.x = av.x; a.y = av.y;
        const float* __restrict__ brow = B + (size_t)(k + 2 * half) * N + l16;
#pragma unroll
        for (int t = 0; t < NTILES; ++t) {
            v2f b;
            b.x = brow[t * 16];             // row K = k+2*half
            b.y = brow[t * 16 + N];         // row K = k+2*half+1
            acc[t] = __builtin_amdgcn_wmma_f32_16x16x4_f32(
                false, a, false, b, (short)0, acc[t], false, false);
        }
    }

#pragma unroll
    for (int t = 0; t < NTILES; ++t) {
        int colc = t * 16 + l16;
        float bv = BIAS ? bias[colc] : 0.0f;
        float* crow = C + (size_t)(row0 + half * 8) * N + colc;
#pragma unroll
        for (int v = 0; v < 8; ++v)
            crow[(size_t)v * N] = acc[t][v] + bv;
    }
}

// ---------------------------------------------------------------------------
// out[i,:] = h[i,:] * dinv[i]^2 + bias   (self-loop message + bias)
// One wave per node row; one float4 per lane covers HID=128.
// ---------------------------------------------------------------------------
__global__ void k_self_bias(const float* __restrict__ h, const float* __restrict__ dinv,
                            const float* __restrict__ bias, float* __restrict__ out, int n) {
    int tid  = blockIdx.x * blockDim.x + threadIdx.x;
    int node = tid >> 5, lane = tid & 31;
    if (node >= n) return;
    float dv = dinv[node];
    float s  = dv * dv;
    float4 hv = ((const float4*)(h + (size_t)node * HID))[lane];
    float4 bv = ((const float4*)bias)[lane];
    float4 o;
    o.x = hv.x * s + bv.x; o.y = hv.y * s + bv.y;
    o.z = hv.z * s + bv.z; o.w = hv.w * s + bv.w;
    ((float4*)(out + (size_t)node * HID))[lane] = o;
}

// ---------------------------------------------------------------------------
// Edge scatter-add: out[c,:] += h[r,:] * dinv[r]*dinv[c]
// One wave per edge: coalesced 512B gather + 128 native f32 atomics.
// ---------------------------------------------------------------------------
__global__ void k_edge_agg(const int* __restrict__ row, const int* __restrict__ col,
                           const float* __restrict__ dinv, const float* __restrict__ h,
                           float* __restrict__ out, int nE) {
    int tid = blockIdx.x * blockDim.x + threadIdx.x;
    int e = tid >> 5, lane = tid & 31;
    if (e >= nE) return;
    int r = row[e], c = col[e];
    float w = dinv[r] * dinv[c];
    float4 hv = ((const float4*)(h + (size_t)r * HID))[lane];
    float* dst = out + (size_t)c * HID + lane * 4;
    atomic_add_f32(dst + 0, hv.x * w);
    atomic_add_f32(dst + 1, hv.y * w);
    atomic_add_f32(dst + 2, hv.z * w);
    atomic_add_f32(dst + 3, hv.w * w);
}

// ---------------------------------------------------------------------------
// Global mean pool: sums[g,:] += relu(out2[i,:]); counts[g] += 1
// ---------------------------------------------------------------------------
__global__ void k_pool(const float* __restrict__ out2, const int* __restrict__ batch,
                       float* __restrict__ sums, float* __restrict__ counts, int n) {
    int tid  = blockIdx.x * blockDim.x + threadIdx.x;
    int node = tid >> 5, lane = tid & 31;
    if (node >= n) return;
    int g = batch[node];
    float4 v = ((const float4*)(out2 + (size_t)node * HID))[lane];
    v.x = fmaxf(v.x, 0.0f); v.y = fmaxf(v.y, 0.0f);
    v.z = fmaxf(v.z, 0.0f); v.w = fmaxf(v.w, 0.0f);
    float* dst = sums + (size_t)g * HID + lane * 4;
    atomic_add_f32(dst + 0, v.x);
    atomic_add_f32(dst + 1, v.y);
    atomic_add_f32(dst + 2, v.z);
    atomic_add_f32(dst + 3, v.w);
    if (lane == 0) atomic_add_f32(&counts[g], 1.0f);
}

__global__ void k_pooled(const float* __restrict__ sums, const float* __restrict__ counts,
                         float* __restrict__ pooled) {
    int i = blockIdx.x * blockDim.x + threadIdx.x;
    if (i < N_GRAPHS * HID) {
        int g = i / HID;
        pooled[i] = sums[i] / fmaxf(counts[g], 1.0f);
    }
}

// ---------------------------------------------------------------------------
extern "C" void kernel_launch(void* const* d_in, const int* in_sizes, int n_in,
                              void* d_out, int out_size, void* d_ws, size_t ws_size,
                              hipStream_t stream) {
    const float* x     = (const float*)d_in[0];
    const int*   ei    = (const int*)d_in[1];     // [2, E] flat
    const int*   batch = (const int*)d_in[2];
    const float* W1    = (const float*)d_in[3];
    const float* b1    = (const float*)d_in[4];
    const float* W2    = (const float*)d_in[5];
    const float* b2    = (const float*)d_in[6];
    const float* Wfc   = (const float*)d_in[7];
    const float* bfc   = (const float*)d_in[8];
    float* out = (float*)d_out;

    const int* row = ei;
    const int* col = ei + N_EDGES;

    char* ws = (char*)d_ws;
    const size_t BIG = (size_t)N_NODES * HID * sizeof(float);     // 51.2 MB
    float* buf1   = (float*)(ws);                 // h (GEMM output)
    float* buf2   = (float*)(ws + BIG);           // aggregated output
    float* deg    = (float*)(ws + 2 * BIG);
    float* dinv   = (float*)(ws + 2 * BIG + 400000);
    float* sums   = (float*)(ws + 2 * BIG + 800000);
    float* counts = (float*)(ws + 2 * BIG + 800000 + N_GRAPHS * HID * 4);
    float* pooled = (float*)(ws + 2 * BIG + 800000 + N_GRAPHS * HID * 4 + 256);

    const int TB = 256;
    // ---- normalization coefficients (shared by both layers) ----
    k_deg_init <<<(N_NODES + TB - 1) / TB, TB, 0, stream>>>(deg, N_NODES);
    k_deg_count<<<(N_EDGES + TB - 1) / TB, TB, 0, stream>>>(col, deg, N_EDGES);
    k_dinv     <<<(N_NODES + TB - 1) / TB, TB, 0, stream>>>(deg, dinv, N_NODES);
    k_zero_pool<<<(N_GRAPHS * HID + TB - 1) / TB, TB, 0, stream>>>(sums, counts);

    const int gemm_threads = (N_NODES / 16) * 32;             // 1 wave / 16 rows
    dim3 ggemm((gemm_threads + TB - 1) / TB);
    dim3 grow (((size_t)N_NODES * 32 + TB - 1) / TB);
    dim3 gedge(((size_t)N_EDGES * 32 + TB - 1) / TB);

    // ---- layer 1: h1 = x @ W1 ; out1 = A_hat h1 + b1 ----
    k_wmma_gemm<HID / 16, false, false><<<ggemm, TB, 0, stream>>>(x, W1, nullptr, buf1, N_NODES, IN_DIM);
    k_self_bias<<<grow, TB, 0, stream>>>(buf1, dinv, b1, buf2, N_NODES);
    k_edge_agg <<<gedge, TB, 0, stream>>>(row, col, dinv, buf1, buf2, N_EDGES);

    // ---- layer 2: h2 = relu(out1) @ W2 ; out2 = A_hat h2 + b2 ----
    k_wmma_gemm<HID / 16, true, false><<<ggemm, TB, 0, stream>>>(buf2, W2, nullptr, buf1, N_NODES, HID);
    k_self_bias<<<grow, TB, 0, stream>>>(buf1, dinv, b2, buf2, N_NODES);
    k_edge_agg <<<gedge, TB, 0, stream>>>(row, col, dinv, buf1, buf2, N_EDGES);

    // ---- global mean pool of relu(out2) ----
    k_pool  <<<grow, TB, 0, stream>>>(buf2, batch, sums, counts, N_NODES);
    k_pooled<<<(N_GRAPHS * HID + TB - 1) / TB, TB, 0, stream>>>(sums, counts, pooled);

    // ---- FC: out = pooled @ Wfc + bfc   (M=64, K=128, N=32) ----
    k_wmma_gemm<OUT_DIM / 16, false, true><<<1, (N_GRAPHS / 16) * 32, 0, stream>>>(
        pooled, Wfc, bfc, out, N_GRAPHS, HID);
    (void)in_sizes; (void)n_in; (void)out_size; (void)ws_size;
}